// GenesisHemisphere_5068061409906
// MI455X (gfx1250) — compile-verified
//
#include <hip/hip_runtime.h>
#include <cmath>
#include <stdint.h>

typedef uint32_t u32;
typedef uint64_t u64;
typedef __attribute__((ext_vector_type(2))) float f32x2;
typedef __attribute__((ext_vector_type(4))) float f32x4;
typedef __attribute__((ext_vector_type(4))) u32   u32x4;
typedef __attribute__((ext_vector_type(8))) u32   u32x8;

#define ROWS_PER_TILE 256
// LDS layout (float offsets within one buffer)
#define OFF_EXT 0      // 256 floats
#define OFF_DZ  256    // 512
#define OFF_DP  768    // 512
#define OFF_TZ  1280   // 768
#define OFF_TP  2048   // 768
#define OFF_QZ  2816   // 1024 (16B aligned: 2816*4=11264)
#define OFF_QP  3840   // 1024 (16B aligned)
#define BUF_FLOATS 4864  // 19456 bytes per buffer (16B multiple)

// ---------------- CDNA5 TDM helpers ----------------

__device__ __forceinline__ void tdm_wait0() {
#if __has_builtin(__builtin_amdgcn_s_wait_tensorcnt)
  __builtin_amdgcn_s_wait_tensorcnt(0);
#else
  asm volatile("s_wait_tensorcnt 0x0" ::: "memory");
#endif
}

// Build D# group1 for a 1-D contiguous copy of `nelem` 4-byte elements.
// Bitfields per CDNA5 ISA.
__device__ __forceinline__ u32x8 tdm_group1(u32 nelem) {
  u32x8 g;
  g[0] = 2u << 16;                        // workgroup_mask=0, data_size=2 (4B)
  g[1] = (nelem & 0xffffu) << 16;         // tensor_dim0[15:0] @ bits 63:48
  g[2] = (nelem >> 16) | (1u << 16);      // tensor_dim0[31:16]; tensor_dim1=1
  g[3] = (nelem & 0xffffu) << 16;         // tensor_dim1 hi=0; tile_dim0=nelem
  g[4] = 0u;                              // tile_dim1=0, tile_dim2=0 (1-D tile)
  g[5] = nelem;                           // tensor_dim0_stride[31:0]
  g[6] = 0u;                              // stride hi, tensor_dim1_stride lo
  g[7] = 0u;
  return g;
}

// Issue one TDM load: global -> LDS. group0 per ISA §8.3.
__device__ __forceinline__ void tdm_issue(u32 lds_byte, const float* gptr,
                                          const u32x8 g1) {
  u32x4 g0;
  u64 ga = (u64)(uintptr_t)gptr;
  g0[0] = 1u;                                            // count=1, user mode
  g0[1] = lds_byte;                                      // lds_addr (bytes)
  g0[2] = (u32)ga;                                       // global_addr[31:0]
  g0[3] = ((u32)(ga >> 32) & 0x01ffffffu) | 0x80000000u; // addr[56:32], type=2
  asm volatile("tensor_load_to_lds %0, %1" :: "s"(g0), "s"(g1) : "memory");
}

// ---------------- per-row DennisNode math ----------------

__device__ __forceinline__ void dennis_row(
    float ext, f32x2 dz, f32x2 dp,
    float tz0, float tz1, float tz2,
    float tp0, float tp1, float tp2,
    f32x4 qz, f32x4 qp,
    const float* tC, const float* qC, float tqc,
    float omd, float omt, float omq, float* o) {
  const float DT = 0.01f, DRIVE = 0.1f, COUPLE = 0.1f, DAMP = 0.01f;
  const float ENTRAIN = 0.1f, EXTS = 0.1f, GAM = 0.103f;

  // D module (unbounded, no entrainment)
  float dpn0 = dp.x + DT * omd;
  float dpn1 = dp.y + DT * omd;
  float s0, c0, s1, c1;
  __sincosf(dpn0, &s0, &c0);
  __sincosf(dpn1, &s1, &c1);
  float ei = EXTS * ext;
  float d0 = dz.x + DT * (DRIVE * s0 + COUPLE * (dz.y + ei) - DAMP * dz.x);
  float d1 = dz.y + DT * (DRIVE * s1 + COUPLE * (dz.x + ei) - DAMP * dz.y);
  float mp = atan2f(s0 + s1, c0 + c1);   // circular mean phase (1/2 cancels)

  // T module (entrained, clamped)
  float tzv[3] = {tz0, tz1, tz2};
  float tpv[3] = {tp0, tp1, tp2};
  float to[3];
#pragma unroll
  for (int i = 0; i < 3; ++i) {
    float nb = tC[3*i]*tzv[0] + tC[3*i+1]*tzv[1] + tC[3*i+2]*tzv[2] + ei;
    float ph = tpv[i] + DT * omt + ENTRAIN * __sinf(mp - tpv[i]);
    float z  = tzv[i] + DT * (DRIVE * __sinf(ph) + COUPLE * nb - DAMP * tzv[i]);
    to[i] = fminf(fmaxf(z, -GAM), GAM);
  }

  // Q module driven by mean T output
  float qi = EXTS * ((to[0] + to[1] + to[2]) * (1.0f / 3.0f) * tqc);
  float qzv[4] = {qz.x, qz.y, qz.z, qz.w};
  float qpv[4] = {qp.x, qp.y, qp.z, qp.w};
  float qo[4];
#pragma unroll
  for (int i = 0; i < 4; ++i) {
    float nb = qC[4*i]*qzv[0] + qC[4*i+1]*qzv[1] + qC[4*i+2]*qzv[2] +
               qC[4*i+3]*qzv[3] + qi;
    float ph = qpv[i] + DT * omq + ENTRAIN * __sinf(mp - qpv[i]);
    float z  = qzv[i] + DT * (DRIVE * __sinf(ph) + COUPLE * nb - DAMP * qzv[i]);
    qo[i] = fminf(fmaxf(z, -GAM), GAM);
  }

  o[0] = d0;    o[1] = d1;
  o[2] = to[0]; o[3] = to[1]; o[4] = to[2];
  o[5] = qo[0]; o[6] = qo[1]; o[7] = qo[2]; o[8] = qo[3];
  o[9] = mp;
}

// Non-temporal streaming store of a 40B row (5 x b64), 8B aligned.
__device__ __forceinline__ void store_row(float* __restrict__ out, u64 row,
                                          const float* o) {
  f32x2* p = (f32x2*)(out + row * 10u);
#pragma unroll
  for (int k = 0; k < 5; ++k) {
    f32x2 v;
    v.x = o[2 * k];
    v.y = o[2 * k + 1];
    __builtin_nontemporal_store(v, p + k);
  }
}

// ---------------- kernel ----------------

__global__ __launch_bounds__(256) void genesis_hemisphere_kernel(
    const float* __restrict__ ext_g, const float* __restrict__ dz_g,
    const float* __restrict__ dp_g,  const float* __restrict__ tz_g,
    const float* __restrict__ tp_g,  const float* __restrict__ qz_g,
    const float* __restrict__ qp_g,  const float* __restrict__ tC_g,
    const float* __restrict__ qC_g,  const float* __restrict__ tqc_g,
    float* __restrict__ out, int n, int tiles,
    float omd, float omt, float omq) {
  __shared__ __align__(16) float smem[2 * BUF_FLOATS];
  const int tid = threadIdx.x;

  // Coupling matrices: uniform addresses -> scalar loads, kept in registers.
  float tC[9], qC[16];
#pragma unroll
  for (int i = 0; i < 9; ++i) tC[i] = tC_g[i];
#pragma unroll
  for (int i = 0; i < 16; ++i) qC[i] = qC_g[i];
  const float tqc = tqc_g[0];

  const u32 lds_base = (u32)(uintptr_t)(&smem[0]);
  const u32x8 g1w1 = tdm_group1(ROWS_PER_TILE * 1);
  const u32x8 g1w2 = tdm_group1(ROWS_PER_TILE * 2);
  const u32x8 g1w3 = tdm_group1(ROWS_PER_TILE * 3);
  const u32x8 g1w4 = tdm_group1(ROWS_PER_TILE * 4);

  // Wave 0 issues TDM. Make the predicate wave-uniform via readfirstlane so
  // the compiler emits a scalar branch (TDM ignores EXEC; pure lane masking
  // would let every wave issue duplicate DMAs).
  const bool issuer =
      ((__builtin_amdgcn_readfirstlane((int)threadIdx.x) >> 5) == 0);

  auto issue_tile = [&](int tt, int pp) {
    const u32 lb = lds_base + (u32)pp * (BUF_FLOATS * 4u);
    const u64 r0 = (u64)tt * ROWS_PER_TILE;
    tdm_issue(lb + OFF_EXT * 4u, ext_g + r0,     g1w1);
    tdm_issue(lb + OFF_DZ  * 4u, dz_g + r0 * 2u, g1w2);
    tdm_issue(lb + OFF_DP  * 4u, dp_g + r0 * 2u, g1w2);
    tdm_issue(lb + OFF_TZ  * 4u, tz_g + r0 * 3u, g1w3);
    tdm_issue(lb + OFF_TP  * 4u, tp_g + r0 * 3u, g1w3);
    tdm_issue(lb + OFF_QZ  * 4u, qz_g + r0 * 4u, g1w4);
    tdm_issue(lb + OFF_QP  * 4u, qp_g + r0 * 4u, g1w4);
  };

  int t = (int)blockIdx.x;
  int p = 0;
  if (issuer && t < tiles) issue_tile(t, 0);

  for (; t < tiles; t += (int)gridDim.x) {
    if (issuer) tdm_wait0();      // tile t resident in buf[p]
    __syncthreads();              // publish LDS; also fences prev compute
    int tn = t + (int)gridDim.x;
    if (issuer && tn < tiles) issue_tile(tn, p ^ 1);  // prefetch next tile

    const float* buf = smem + p * BUF_FLOATS;
    float ext = buf[OFF_EXT + tid];
    f32x2 dz = *(const f32x2*)(buf + OFF_DZ + 2 * tid);
    f32x2 dp = *(const f32x2*)(buf + OFF_DP + 2 * tid);
    float tz0 = buf[OFF_TZ + 3 * tid + 0];
    float tz1 = buf[OFF_TZ + 3 * tid + 1];
    float tz2 = buf[OFF_TZ + 3 * tid + 2];
    float tp0 = buf[OFF_TP + 3 * tid + 0];
    float tp1 = buf[OFF_TP + 3 * tid + 1];
    float tp2 = buf[OFF_TP + 3 * tid + 2];
    f32x4 qz = *(const f32x4*)(buf + OFF_QZ + 4 * tid);
    f32x4 qp = *(const f32x4*)(buf + OFF_QP + 4 * tid);

    float o[10];
    dennis_row(ext, dz, dp, tz0, tz1, tz2, tp0, tp1, tp2, qz, qp,
               tC, qC, tqc, omd, omt, omq, o);
    store_row(out, (u64)t * ROWS_PER_TILE + tid, o);
    p ^= 1;
  }

  // Tail rows (B % 256), straight from global memory.
  if (blockIdx.x == 0) {
    for (int i = tiles * ROWS_PER_TILE + tid; i < n; i += ROWS_PER_TILE) {
      float ext = ext_g[i];
      f32x2 dz = *(const f32x2*)(dz_g + 2 * (u64)i);
      f32x2 dp = *(const f32x2*)(dp_g + 2 * (u64)i);
      float tz0 = tz_g[3 * (u64)i + 0], tz1 = tz_g[3 * (u64)i + 1],
            tz2 = tz_g[3 * (u64)i + 2];
      float tp0 = tp_g[3 * (u64)i + 0], tp1 = tp_g[3 * (u64)i + 1],
            tp2 = tp_g[3 * (u64)i + 2];
      f32x4 qz = *(const f32x4*)(qz_g + 4 * (u64)i);
      f32x4 qp = *(const f32x4*)(qp_g + 4 * (u64)i);
      float o[10];
      dennis_row(ext, dz, dp, tz0, tz1, tz2, tp0, tp1, tp2, qz, qp,
                 tC, qC, tqc, omd, omt, omq, o);
      store_row(out, (u64)i, o);
    }
  }
}

// ---------------- launch ----------------

extern "C" void kernel_launch(void* const* d_in, const int* in_sizes, int n_in,
                              void* d_out, int out_size, void* d_ws,
                              size_t ws_size, hipStream_t stream) {
  const float* ext = (const float*)d_in[0];
  const float* dz  = (const float*)d_in[1];
  const float* dp  = (const float*)d_in[2];
  const float* tz  = (const float*)d_in[3];
  const float* tp  = (const float*)d_in[4];
  const float* qz  = (const float*)d_in[5];
  const float* qp  = (const float*)d_in[6];
  const float* tC  = (const float*)d_in[7];
  const float* qC  = (const float*)d_in[8];
  const float* tqc = (const float*)d_in[9];

  const int n = in_sizes[0];
  const int tiles = n / ROWS_PER_TILE;

  const double PHI = 1.618033988749895;
  const double TWO_PI = 6.283185307179586;
  const float omd = (float)(TWO_PI / (40.0 * std::pow(PHI, 11.0)));
  const float omt = (float)(TWO_PI / (40.0 * std::pow(PHI, 7.0)));
  const float omq = (float)(TWO_PI / (40.0 * std::pow(PHI, 5.0)));

  int grid = tiles;
  if (grid > 2048) grid = 2048;   // persistent blocks, double-buffered tiles
  if (grid < 1) grid = 1;

  genesis_hemisphere_kernel<<<dim3(grid), dim3(256), 0, stream>>>(
      ext, dz, dp, tz, tp, qz, qp, tC, qC, tqc, (float*)d_out, n, tiles,
      omd, omt, omq);
}